// KMeans_2757369004654
// MI455X (gfx1250) — compile-verified
//
#include <hip/hip_runtime.h>
#include <hip/hip_bf16.h>

#define B_ 16
#define N_ 32768
#define D_ 64
#define K_ 128
#define ITERS_ 10

#define BLOCKS_PER_BATCH 32
#define PTS_PER_BLOCK (N_ / BLOCKS_PER_BATCH)   // 1024
#define CENT_STRIDE 65                          // padded LDS row (floats), conflict-free

typedef float v2f __attribute__((ext_vector_type(2)));
typedef float v8f __attribute__((ext_vector_type(8)));

// ---------------------------------------------------------------------------
// init: cent = x[:, :K, :], sums = 0, cnts = 0
// ---------------------------------------------------------------------------
__global__ void kmeans_init(const float* __restrict__ x,
                            float* __restrict__ cent,
                            float* __restrict__ sums,
                            float* __restrict__ cnts) {
    int tid = blockIdx.x * blockDim.x + threadIdx.x;
    if (tid < B_ * K_ * D_) {
        int b   = tid / (K_ * D_);
        int rem = tid % (K_ * D_);
        cent[tid] = x[(size_t)b * N_ * D_ + rem];
        sums[tid] = 0.0f;
    }
    if (tid < B_ * K_) cnts[tid] = 0.0f;
}

// ---------------------------------------------------------------------------
// assign: per point, argmin_k (|c_k|^2 - 2 x·c_k); accumulate sums/counts.
// Grid: B_*BLOCKS_PER_BATCH blocks of 256 threads (8 wave32).
// Dynamic LDS: centLDS[K*65] | csqLDS[K] | sumLDS[K*64] | cntLDS[K]
// ---------------------------------------------------------------------------
__launch_bounds__(256)
__global__ void kmeans_assign(const float* __restrict__ x,
                              const float* __restrict__ cent,
                              float* __restrict__ sums,
                              float* __restrict__ cnts,
                              int* __restrict__ idxOut) {
    extern __shared__ float smem[];
    float* centLDS = smem;                              // K_*CENT_STRIDE
    float* csqLDS  = centLDS + K_ * CENT_STRIDE;        // K_
    float* sumLDS  = csqLDS + K_;                       // K_*D_
    float* cntLDS  = sumLDS + K_ * D_;                  // K_

    const int b   = blockIdx.x / BLOCKS_PER_BATCH;
    const int blk = blockIdx.x % BLOCKS_PER_BATCH;

    const float* xb = x + (size_t)b * N_ * D_;
    const float* cb = cent + (size_t)b * K_ * D_;

    // cooperative load of centroids (padded) + zero partial accumulators
    for (int i = threadIdx.x; i < K_ * D_; i += 256) {
        int k = i >> 6, d = i & 63;
        centLDS[k * CENT_STRIDE + d] = cb[i];
        sumLDS[i] = 0.0f;
    }
    if (threadIdx.x < K_) cntLDS[threadIdx.x] = 0.0f;
    __syncthreads();

    if (threadIdx.x < K_) {
        float s = 0.0f;
        #pragma unroll
        for (int d = 0; d < D_; ++d) {
            float c = centLDS[threadIdx.x * CENT_STRIDE + d];
            s += c * c;
        }
        csqLDS[threadIdx.x] = s;
    }
    __syncthreads();

    const int wave = threadIdx.x >> 5;   // 0..7
    const int lane = threadIdx.x & 31;
    const int col  = lane & 15;          // M row (A) / N column (B)
    const int hi   = lane >> 4;          // K-pair selector within fragments

    for (int pass = 0; pass < PTS_PER_BLOCK / 128; ++pass) {
        const int pbase = blk * PTS_PER_BLOCK + pass * 128 + wave * 16;

        // --- preload B fragments: 16 points x 64 dims, 16 chunks of 4x16 f32
        // lane layout (assumed, mirrors documented A layout): lane=col(+16*hi),
        // v0 = K=2*hi, v1 = K=2*hi+1 within each 4-wide chunk.
        v2f bfrag[16];
        const float* prow = xb + (size_t)(pbase + col) * D_;
        #pragma unroll
        for (int c = 0; c < 16; ++c) {
            int d0 = 4 * c + 2 * hi;
            bfrag[c].x = prow[d0];
            bfrag[c].y = prow[d0 + 1];
        }

        float bestVal = 3.4e38f;
        int   bestIdx = 0;

        #pragma unroll
        for (int mt = 0; mt < 8; ++mt) {             // 8 tiles of 16 centroids
            v8f acc = {};
            #pragma unroll
            for (int c = 0; c < 16; ++c) {           // D = 64 in chunks of 4
                int d0 = 4 * c + 2 * hi;
                const float* arow = &centLDS[(mt * 16 + col) * CENT_STRIDE + d0];
                v2f afrag;
                afrag.x = arow[0];
                afrag.y = arow[1];
                acc = __builtin_amdgcn_wmma_f32_16x16x4_f32(
                        /*neg_a=*/false, afrag, /*neg_b=*/false, bfrag[c],
                        /*c_mod=*/(short)0, acc, /*reuse_a=*/false, /*reuse_b=*/false);
            }
            // score rows: lane holds point 'col', centroids m = mt*16 + 8*hi + r
            #pragma unroll
            for (int r = 0; r < 8; ++r) {
                int   m = mt * 16 + 8 * hi + r;
                float s = csqLDS[m] - 2.0f * acc[r];
                if (s < bestVal) { bestVal = s; bestIdx = m; }
            }
        }

        // merge the (lane, lane+16) halves of each point column
        float oVal = __shfl_xor(bestVal, 16, 32);
        int   oIdx = __shfl_xor(bestIdx, 16, 32);
        if (oVal < bestVal || (oVal == bestVal && oIdx < bestIdx)) {
            bestVal = oVal; bestIdx = oIdx;
        }

        if (hi == 0) idxOut[(size_t)b * N_ + pbase + col] = bestIdx;

        // accumulate the 16 points into LDS partial sums (ds_add_f32)
        #pragma unroll
        for (int p = 0; p < 16; ++p) {
            int kbest = __shfl(bestIdx, p, 32);
            const float* pr = xb + (size_t)(pbase + p) * D_;
            atomicAdd(&sumLDS[kbest * D_ + lane],      pr[lane]);
            atomicAdd(&sumLDS[kbest * D_ + lane + 32], pr[lane + 32]);
            if (lane == 0) atomicAdd(&cntLDS[kbest], 1.0f);
        }
    }

    __syncthreads();
    // flush block partials to global (global_atomic_add_f32)
    float* sb = sums + (size_t)b * K_ * D_;
    for (int i = threadIdx.x; i < K_ * D_; i += 256) atomicAdd(&sb[i], sumLDS[i]);
    if (threadIdx.x < K_) atomicAdd(&cnts[b * K_ + threadIdx.x], cntLDS[threadIdx.x]);
}

// ---------------------------------------------------------------------------
// update: cent = cnt>0 ? sums/max(cnt,1) : cent ; re-zero sums/cnts
// Grid: B_*K_ blocks of 64 threads
// ---------------------------------------------------------------------------
__global__ void kmeans_update(float* __restrict__ cent,
                              float* __restrict__ sums,
                              float* __restrict__ cnts) {
    int bk = blockIdx.x;
    int d  = threadIdx.x;
    float cnt = cnts[bk];
    float sv  = sums[(size_t)bk * D_ + d];
    float cv  = cent[(size_t)bk * D_ + d];
    __syncthreads();
    cent[(size_t)bk * D_ + d] = (cnt > 0.0f) ? (sv / fmaxf(cnt, 1.0f)) : cv;
    sums[(size_t)bk * D_ + d] = 0.0f;
    if (d == 0) cnts[bk] = 0.0f;
}

// ---------------------------------------------------------------------------
extern "C" void kernel_launch(void* const* d_in, const int* in_sizes, int n_in,
                              void* d_out, int out_size, void* d_ws, size_t ws_size,
                              hipStream_t stream) {
    const float* x = (const float*)d_in[0];

    float* cent = (float*)d_ws;                 // B*K*D floats
    float* sums = cent + B_ * K_ * D_;          // B*K*D floats
    float* cnts = sums + B_ * K_ * D_;          // B*K  floats

    int*   idxOut  = (int*)d_out;               // B*N int32 (bit-stored)
    float* centOut = (float*)d_out + (size_t)B_ * N_;

    kmeans_init<<<(B_ * K_ * D_ + 255) / 256, 256, 0, stream>>>(x, cent, sums, cnts);

    const size_t ldsBytes =
        (size_t)(K_ * CENT_STRIDE + K_ + K_ * D_ + K_) * sizeof(float);  // ~67 KB

    for (int it = 0; it < ITERS_; ++it) {
        kmeans_assign<<<B_ * BLOCKS_PER_BATCH, 256, ldsBytes, stream>>>(
            x, cent, sums, cnts, idxOut);
        kmeans_update<<<B_ * K_, 64, 0, stream>>>(cent, sums, cnts);
    }

    hipMemcpyAsync(centOut, cent, (size_t)B_ * K_ * D_ * sizeof(float),
                   hipMemcpyDeviceToDevice, stream);
}